// BPDecoder_69441031242297
// MI455X (gfx1250) — compile-verified
//
#include <hip/hip_runtime.h>
#include <math.h>

typedef float v2f __attribute__((ext_vector_type(2)));
typedef float v8f __attribute__((ext_vector_type(8)));

#define J_DIM   8
#define N_DIM   2048
#define D_DIM   64
#define TILE_M  128
#define TILE_N  64
#define LDS_STRIDE 68   // 64 + 4 pad: 16B-aligned rows, conflict-free frag reads

__global__ __launch_bounds__(256)
void bp_decoder_gemm(const float* __restrict__ Z,
                     const float* __restrict__ rk_logit,
                     float* __restrict__ out)
{
    __shared__ float As[TILE_M * LDS_STRIDE];   // rk-scaled rows (A = Z*rk)
    __shared__ float Bs[TILE_N * LDS_STRIDE];   // raw rows (B = Z^T columns)
    __shared__ float rk_s[D_DIM];

    const int t  = threadIdx.x;
    const int j  = blockIdx.z;
    const int m0 = blockIdx.y * TILE_M;
    const int n0 = blockIdx.x * TILE_N;

    if (t < D_DIM) {
        float l = rk_logit[t];
        rk_s[t] = 1.0f / (1.0f + __expf(-l));
    }
    __syncthreads();

    const float* __restrict__ Zj = Z + (size_t)j * N_DIM * D_DIM;

    // Stage A tile: 128 rows x 64 cols, scaled by rk. 2048 float4 units / 256 thr = 8 iters.
    #pragma unroll
    for (int it = 0; it < 8; ++it) {
        int idx = it * 256 + t;
        int row = idx >> 4;           // /16 float4s per row
        int c4  = (idx & 15) << 2;
        float4 za = *(const float4*)(Zj + (size_t)(m0 + row) * D_DIM + c4);
        float4 sa;
        sa.x = za.x * rk_s[c4 + 0];
        sa.y = za.y * rk_s[c4 + 1];
        sa.z = za.z * rk_s[c4 + 2];
        sa.w = za.w * rk_s[c4 + 3];
        *(float4*)(&As[row * LDS_STRIDE + c4]) = sa;
    }
    // Stage B tile: 64 rows x 64 cols, unscaled. 1024 float4 units / 256 thr = 4 iters.
    #pragma unroll
    for (int it = 0; it < 4; ++it) {
        int idx = it * 256 + t;
        int row = idx >> 4;
        int c4  = (idx & 15) << 2;
        float4 zb = *(const float4*)(Zj + (size_t)(n0 + row) * D_DIM + c4);
        *(float4*)(&Bs[row * LDS_STRIDE + c4]) = zb;
    }
    __syncthreads();

    const int wave  = t >> 5;
    const int lane  = t & 31;
    const int lrow  = lane & 15;            // M (for A) / N (for B) within tile
    const int kofs  = (lane >> 4) << 1;     // K-pair offset: 0 (lanes 0-15) or 2 (16-31)
    const int m_sub = wave * 16;            // this wave's 16-row strip

    // Preload all 16 A fragments for this wave's row tile (32 VGPRs).
    v2f afrag[16];
    #pragma unroll
    for (int kk = 0; kk < 16; ++kk) {
        afrag[kk] = *(const v2f*)(&As[(m_sub + lrow) * LDS_STRIDE + kk * 4 + kofs]);
    }

    float* __restrict__ outj = out + (size_t)j * N_DIM * N_DIM;
    const int row_hi = (lane >> 4) * 8;     // C/D layout: VGPR v -> M = v + 8*(lane/16)

    #pragma unroll
    for (int nt = 0; nt < 4; ++nt) {
        const int n_sub = nt * 16;
        v8f acc = {};
        #pragma unroll
        for (int kk = 0; kk < 16; ++kk) {
            v2f b = *(const v2f*)(&Bs[(n_sub + lrow) * LDS_STRIDE + kk * 4 + kofs]);
            // D = A x B + C ; emits v_wmma_f32_16x16x4_f32
            acc = __builtin_amdgcn_wmma_f32_16x16x4_f32(
                /*neg_a=*/false, afrag[kk],
                /*neg_b=*/false, b,
                /*c_mod=*/(short)0, acc,
                /*reuse_a=*/false, /*reuse_b=*/false);
        }
        // Epilogue: clamp + complementary log-log link, store 16x16 tile.
        #pragma unroll
        for (int v = 0; v < 8; ++v) {
            float x = acc[v];
            x = fminf(x, 10.0f);
            float y = 1.0f - __expf(-__expf(x));
            int gm = m0 + m_sub + v + row_hi;
            int gn = n0 + n_sub + lrow;
            outj[(size_t)gm * N_DIM + gn] = y;
        }
    }
}

// Copies Z (pass-through output #2) and writes rk = sigmoid(rk_logit) (output #3).
__global__ __launch_bounds__(256)
void bp_decoder_tail(const float* __restrict__ Z,
                     const float* __restrict__ rk_logit,
                     float* __restrict__ out)
{
    const size_t adjElems = (size_t)J_DIM * N_DIM * N_DIM;   // 33,554,432
    const size_t zElems   = (size_t)J_DIM * N_DIM * D_DIM;   // 1,048,576

    size_t i = (size_t)blockIdx.x * 256 + threadIdx.x;       // float4 index over Z
    const float4* src = (const float4*)Z;
    float4* dst = (float4*)(out + adjElems);
    dst[i] = src[i];

    if (blockIdx.x == 0 && threadIdx.x < D_DIM) {
        float l = rk_logit[threadIdx.x];
        out[adjElems + zElems + threadIdx.x] = 1.0f / (1.0f + __expf(-l));
    }
}

extern "C" void kernel_launch(void* const* d_in, const int* in_sizes, int n_in,
                              void* d_out, int out_size, void* d_ws, size_t ws_size,
                              hipStream_t stream) {
    (void)in_sizes; (void)n_in; (void)out_size; (void)d_ws; (void)ws_size;
    const float* Z        = (const float*)d_in[0];
    const float* rk_logit = (const float*)d_in[1];
    float* out = (float*)d_out;

    dim3 grid(N_DIM / TILE_N, N_DIM / TILE_M, J_DIM);   // (32, 16, 8)
    bp_decoder_gemm<<<grid, 256, 0, stream>>>(Z, rk_logit, out);

    // Z copy: 1,048,576 floats = 262,144 float4 -> 1024 blocks of 256.
    bp_decoder_tail<<<1024, 256, 0, stream>>>(Z, rk_logit, out);
}